// GAT_model_10453950399127
// MI455X (gfx1250) — compile-verified
//
#include <hip/hip_runtime.h>
#include <hip/hip_bf16.h>
#include <math.h>

typedef __attribute__((ext_vector_type(2))) float v2f;
typedef __attribute__((ext_vector_type(8))) float v8f;

#define HEADS 4
#define NEG_SLOPE 0.2f
#define ENC_NEG_INF 0x007FFFFFu   // enc(-inf)

// ---- monotonic float<->uint encoding for atomic max over signed floats ----
__device__ __forceinline__ unsigned enc_f32(float f) {
    unsigned u = __float_as_uint(f);
    return (u & 0x80000000u) ? ~u : (u | 0x80000000u);
}
__device__ __forceinline__ float dec_f32(unsigned e) {
    unsigned u = (e & 0x80000000u) ? (e & 0x7FFFFFFFu) : ~e;
    return __uint_as_float(u);
}

// ---------------------------------------------------------------------------
// C[M,Nc] = A[M,K] @ B[K,Nc], fp32, via V_WMMA_F32_16X16X4_F32.
// One wave computes a 64x16 output strip (4 row-tiles sharing the B fragment):
// per k-step = 4 x b64 (A) + 2 x b32 (B) loads for 4 WMMAs.
// Requires Nc%16==0, K%4==0, M%16==0 (partial 64-row block handled by
// clamped loads + wave-uniform store guards; EXEC stays all-ones for WMMA).
// A-frag (16x4 f32): lane half h, lm=lane&15 -> A[row+lm, k+2h+{0,1}]
// B-frag (4x16 f32): lane -> B[k+2h+{0,1}, col0+lm]
// C/D  (16x16 f32): vgpr r, half h -> C[row + r + 8h, col0+lm]
// ---------------------------------------------------------------------------
__global__ __launch_bounds__(256)
void gemm_wmma_f32_rb4(const float* __restrict__ A, const float* __restrict__ B,
                       float* __restrict__ C, int M, int K, int Nc) {
    const int lane  = threadIdx.x & 31;
    const int wave  = threadIdx.x >> 5;
    const int tilesN = Nc >> 4;
    const int rowBlocks = (M + 63) >> 6;
    const int tile = blockIdx.x * (blockDim.x >> 5) + wave;
    if (tile >= rowBlocks * tilesN) return;          // wave-uniform exit
    const int row0 = (tile / tilesN) << 6;
    const int col0 = (tile % tilesN) << 4;
    const int half = lane >> 4;
    const int lm   = lane & 15;

    // clamped per-subtile load rows (partial last block reads row M-1; stores guarded)
    int r0 = row0      + lm; if (r0 > M - 1) r0 = M - 1;
    int r1 = row0 + 16 + lm; if (r1 > M - 1) r1 = M - 1;
    int r2 = row0 + 32 + lm; if (r2 > M - 1) r2 = M - 1;
    int r3 = row0 + 48 + lm; if (r3 > M - 1) r3 = M - 1;

    v8f acc0 = {0.f,0.f,0.f,0.f,0.f,0.f,0.f,0.f};
    v8f acc1 = acc0, acc2 = acc0, acc3 = acc0;

    const float* ap0 = A + (size_t)r0 * K + 2 * half;   // 8B aligned (k+2h even)
    const float* ap1 = A + (size_t)r1 * K + 2 * half;
    const float* ap2 = A + (size_t)r2 * K + 2 * half;
    const float* ap3 = A + (size_t)r3 * K + 2 * half;
    const float* bp  = B + (size_t)(2 * half) * Nc + col0 + lm;

    for (int k = 0; k < K; k += 4) {
        v2f b;
        b.x = bp[0];                                    // K = 2h+0
        b.y = bp[Nc];                                   // K = 2h+1
        v2f a0 = *reinterpret_cast<const v2f*>(ap0);
        v2f a1 = *reinterpret_cast<const v2f*>(ap1);
        v2f a2 = *reinterpret_cast<const v2f*>(ap2);
        v2f a3 = *reinterpret_cast<const v2f*>(ap3);
        acc0 = __builtin_amdgcn_wmma_f32_16x16x4_f32(false, a0, false, b, (short)0, acc0, false, false);
        acc1 = __builtin_amdgcn_wmma_f32_16x16x4_f32(false, a1, false, b, (short)0, acc1, false, false);
        acc2 = __builtin_amdgcn_wmma_f32_16x16x4_f32(false, a2, false, b, (short)0, acc2, false, false);
        acc3 = __builtin_amdgcn_wmma_f32_16x16x4_f32(false, a3, false, b, (short)0, acc3, false, false);
        ap0 += 4; ap1 += 4; ap2 += 4; ap3 += 4;
        bp  += (size_t)4 * Nc;
    }

    const int colIdx = col0 + lm;
    {   // subtile stores (wave-uniform guards; M%16==0 so a subtile is all-or-nothing)
        if (row0 < M) {
            float* cp = C + (size_t)(row0 + 8 * half) * Nc + colIdx;
#pragma unroll
            for (int r = 0; r < 8; ++r) cp[(size_t)r * Nc] = acc0[r];
        }
        if (row0 + 16 < M) {
            float* cp = C + (size_t)(row0 + 16 + 8 * half) * Nc + colIdx;
#pragma unroll
            for (int r = 0; r < 8; ++r) cp[(size_t)r * Nc] = acc1[r];
        }
        if (row0 + 32 < M) {
            float* cp = C + (size_t)(row0 + 32 + 8 * half) * Nc + colIdx;
#pragma unroll
            for (int r = 0; r < 8; ++r) cp[(size_t)r * Nc] = acc2[r];
        }
        if (row0 + 48 < M) {
            float* cp = C + (size_t)(row0 + 48 + 8 * half) * Nc + colIdx;
#pragma unroll
            for (int r = 0; r < 8; ++r) cp[(size_t)r * Nc] = acc3[r];
        }
    }
}

// el[n,h] = sum_d ft[n,h,d]*al[h,d]; er likewise.
__global__ void attn_scores(const float* __restrict__ ft, const float* __restrict__ al,
                            const float* __restrict__ ar, float* __restrict__ el,
                            float* __restrict__ er, int N, int H, int D) {
    int i = blockIdx.x * blockDim.x + threadIdx.x;
    if (i >= N * H) return;
    int h = i % H;
    const float* f = ft + (size_t)i * D;
    const float* l = al + (size_t)h * D;
    const float* r = ar + (size_t)h * D;
    float sl = 0.f, sr = 0.f;
    for (int d = 0; d < D; ++d) { float v = f[d]; sl += v * l[d]; sr += v * r[d]; }
    el[i] = sl; er[i] = sr;
}

__global__ void fill_u32(unsigned* __restrict__ p, unsigned v, size_t n) {
    size_t i = (size_t)blockIdx.x * blockDim.x + threadIdx.x;
    if (i < n) p[i] = v;
}

// e = leaky_relu(el[src]+er[dst]); stash e; segment-max into emax (encoded).
__global__ void edge_pass1(const float* __restrict__ el, const float* __restrict__ er,
                           const int* __restrict__ src, const int* __restrict__ dst,
                           float* __restrict__ ee, unsigned* __restrict__ emax,
                           int E, int H) {
    int i = blockIdx.x * blockDim.x + threadIdx.x;
    if (i >= E * H) return;
    int e = i / H, h = i % H;
    int s = src[e], d = dst[e];
    float x = el[(size_t)s * H + h] + er[(size_t)d * H + h];
    x = x > 0.f ? x : NEG_SLOPE * x;
    ee[i] = x;
    atomicMax(&emax[(size_t)d * H + h], enc_f32(x));
}

// ex = exp(e - emax[dst]); stash ex (overwrite e); segment-sum into denom.
__global__ void edge_pass2(float* __restrict__ ee, const unsigned* __restrict__ emax,
                           const int* __restrict__ dst, float* __restrict__ denom,
                           int E, int H) {
    int i = blockIdx.x * blockDim.x + threadIdx.x;
    if (i >= E * H) return;
    int e = i / H, h = i % H;
    float m = dec_f32(emax[(size_t)dst[e] * H + h]);
    if (!isfinite(m)) m = 0.f;
    float ex = expf(ee[i] - m);
    ee[i] = ex;
    atomicAdd(&denom[(size_t)dst[e] * H + h], ex);
}

// One wave per (edge, head): agg[dst,h,:] += (ex/denom[dst,h]) * ft[src,h,:]
__global__ __launch_bounds__(256)
void edge_aggregate(const float* __restrict__ ee, const float* __restrict__ denom,
                    const int* __restrict__ src, const int* __restrict__ dst,
                    const float* __restrict__ ft, float* __restrict__ agg,
                    int E, int H, int D) {
    int wid  = (blockIdx.x * blockDim.x + threadIdx.x) >> 5;
    int lane = threadIdx.x & 31;
    if (wid >= E * H) return;
    int e = wid / H, h = wid % H;
    int s = src[e], d = dst[e];
    float den   = denom[(size_t)d * H + h];
    float alpha = ee[wid] / fmaxf(den, 1e-9f);
    const float* fsrc = ft  + ((size_t)s * H + h) * D;
    float*       adst = agg + ((size_t)d * H + h) * D;
    for (int i = lane; i < D; i += 32)
        atomicAdd(&adst[i], alpha * fsrc[i]);
}

__global__ void elu_inplace(float* __restrict__ x, size_t n) {
    size_t i = (size_t)blockIdx.x * blockDim.x + threadIdx.x;
    if (i >= n) return;
    float v = x[i];
    x[i] = v > 0.f ? v : expm1f(v);
}

// out[n,c] = mean_h agg[n,h,c]
__global__ void head_mean(const float* __restrict__ agg, float* __restrict__ out,
                          int N, int H, int C) {
    int i = blockIdx.x * blockDim.x + threadIdx.x;
    if (i >= N * C) return;
    int n = i / C, c = i % C;
    float s = 0.f;
    for (int h = 0; h < H; ++h) s += agg[((size_t)n * H + h) * C + c];
    out[i] = s / (float)H;
}

__global__ void log_softmax_rows(float* __restrict__ x, int N, int C) {
    int n = blockIdx.x * blockDim.x + threadIdx.x;
    if (n >= N) return;
    float* row = x + (size_t)n * C;
    float m = -INFINITY;
    for (int c = 0; c < C; ++c) m = fmaxf(m, row[c]);
    float s = 0.f;
    for (int c = 0; c < C; ++c) s += expf(row[c] - m);
    float lse = m + logf(s);
    for (int c = 0; c < C; ++c) row[c] -= lse;
}

extern "C" void kernel_launch(void* const* d_in, const int* in_sizes, int n_in,
                              void* d_out, int out_size, void* d_ws, size_t ws_size,
                              hipStream_t stream) {
    const float* features = (const float*)d_in[0];
    const int*   src      = (const int*)d_in[1];
    const int*   dst      = (const int*)d_in[2];
    const float* W0  = (const float*)d_in[3];
    const float* al0 = (const float*)d_in[4];
    const float* ar0 = (const float*)d_in[5];
    const float* W1  = (const float*)d_in[6];
    const float* al1 = (const float*)d_in[7];
    const float* ar1 = (const float*)d_in[8];
    const float* W2  = (const float*)d_in[9];
    const float* al2 = (const float*)d_in[10];
    const float* ar2 = (const float*)d_in[11];

    const int IN_FEATS = 256, HIDDEN = 64, CLASSES = 40;
    const int N = in_sizes[0] / IN_FEATS;        // 50000
    const int E = in_sizes[1];                   // 800000
    float* out = (float*)d_out;

    // ---- carve workspace (256B aligned chunks) ----
    char* w = (char*)d_ws;
    auto carve = [&](size_t bytes) {
        void* p = (void*)w;
        w += (bytes + 255) & ~(size_t)255;
        return p;
    };
    float*    ft    = (float*)   carve((size_t)N * 256 * sizeof(float));
    float*    hb    = (float*)   carve((size_t)N * 256 * sizeof(float));  // agg / hidden
    float*    ee    = (float*)   carve((size_t)E * HEADS * sizeof(float));
    float*    el    = (float*)   carve((size_t)N * HEADS * sizeof(float));
    float*    er    = (float*)   carve((size_t)N * HEADS * sizeof(float));
    unsigned* emax  = (unsigned*)carve((size_t)N * HEADS * sizeof(unsigned));
    float*    denom = (float*)   carve((size_t)N * HEADS * sizeof(float));
    (void)ws_size;

    auto layer = [&](const float* hin, const float* W, const float* al, const float* ar,
                     int K, int D, float* agg, bool act) {
        const int F = HEADS * D;
        // 1. ft = hin @ W  (WMMA, 64x16 strip per wave)
        const int tiles = ((N + 63) / 64) * (F / 16);
        gemm_wmma_f32_rb4<<<(tiles + 7) / 8, 256, 0, stream>>>(hin, W, ft, N, K, F);
        // 2. attention logits
        const int nh = N * HEADS;
        attn_scores<<<(nh + 255) / 256, 256, 0, stream>>>(ft, al, ar, el, er, N, HEADS, D);
        // 3. init segment buffers + output accumulator
        fill_u32<<<(nh + 255) / 256, 256, 0, stream>>>(emax, ENC_NEG_INF, (size_t)nh);
        fill_u32<<<(nh + 255) / 256, 256, 0, stream>>>((unsigned*)denom, 0u, (size_t)nh);
        const size_t nf = (size_t)N * F;
        fill_u32<<<((unsigned)((nf + 255) / 256)), 256, 0, stream>>>((unsigned*)agg, 0u, nf);
        // 4-5. edge softmax (max, then exp+sum)
        const int eh = E * HEADS;
        edge_pass1<<<(eh + 255) / 256, 256, 0, stream>>>(el, er, src, dst, ee, emax, E, HEADS);
        edge_pass2<<<(eh + 255) / 256, 256, 0, stream>>>(ee, emax, dst, denom, E, HEADS);
        // 6. weighted aggregation: one wave per (edge, head)
        edge_aggregate<<<(eh + 7) / 8, 256, 0, stream>>>(ee, denom, src, dst, ft, agg, E, HEADS, D);
        // 7. activation
        if (act) elu_inplace<<<((unsigned)((nf + 255) / 256)), 256, 0, stream>>>(agg, nf);
    };

    // Layer 0: features -> hb ; Layers 1/2: hb -> hb (GEMM reads hb before fill/scatter)
    layer(features, W0, al0, ar0, IN_FEATS,       HIDDEN,  hb, true);
    layer(hb,       W1, al1, ar1, HEADS * HIDDEN, HIDDEN,  hb, true);
    layer(hb,       W2, al2, ar2, HEADS * HIDDEN, CLASSES, hb, false);

    // mean over heads -> d_out, then in-place log_softmax
    const int nc = N * CLASSES;
    head_mean<<<(nc + 255) / 256, 256, 0, stream>>>(hb, out, N, HEADS, CLASSES);
    log_softmax_rows<<<(N + 255) / 256, 256, 0, stream>>>(out, N, CLASSES);
}